// GNNLayer_77850577207791
// MI455X (gfx1250) — compile-verified
//
#include <hip/hip_runtime.h>
#include <hip/hip_bf16.h>

typedef __attribute__((ext_vector_type(2))) float v2f;
typedef __attribute__((ext_vector_type(8))) float v8f;

#define FEATS 128

// Native f32 global atomic add (no return) -> global_atomic_add_f32, STOREcnt tracked.
__device__ __forceinline__ void gatomic_add(float* p, float v) {
    asm volatile("global_atomic_add_f32 %0, %1, off" :: "v"(p), "v"(v) : "memory");
}

// ---------------------------------------------------------------------------
// xw = x @ W  using V_WMMA_F32_16X16X4_F32.
// Block = 256 threads (8 waves). blockIdx.x = M-tile (16 rows). Wave w owns
// N-tile [16w, 16w+16). W (128x128 f32, 64KB) + 16 rows of x (8KB) staged in LDS.
// A frag (16x4 f32, 2 VGPR): lanes 0-15 hold K=k,k+1 of row M=lane;
//                            lanes 16-31 hold K=k+2,k+3 of row M=lane-16.
// B frag (4x16 f32, 2 VGPR): VGPR0 lanes0-15 = row K=k,   lanes16-31 = row K=k+1;
//                            VGPR1 lanes0-15 = row K=k+2, lanes16-31 = row K=k+3.
// D (16x16 f32, 8 VGPR): VGPR r -> (M=r, N=lane) / (M=r+8, N=lane-16).
// ---------------------------------------------------------------------------
__global__ __launch_bounds__(256) void gemm_xw(const float* __restrict__ x,
                                               const float* __restrict__ W,
                                               float* __restrict__ xw, int n) {
    __shared__ float sW[FEATS * FEATS];   // 64 KB
    __shared__ float sX[16 * FEATS];      // 8 KB
    const int tid = threadIdx.x;
    const int m0  = blockIdx.x * 16;

    for (int i = tid; i < FEATS * FEATS / 4; i += 256)
        ((float4*)sW)[i] = ((const float4*)W)[i];
    for (int i = tid; i < 16 * FEATS / 4; i += 256) {
        int r = i >> 5, c = i & 31;
        int row = m0 + r;
        float4 v = make_float4(0.f, 0.f, 0.f, 0.f);
        if (row < n) v = ((const float4*)(x + (size_t)row * FEATS))[c];
        ((float4*)sX)[i] = v;
    }
    __syncthreads();

    const int wave = tid >> 5;
    const int lane = tid & 31;
    const int n0   = wave * 16;
    const int lrow = lane & 15;
    const int hi   = lane >> 4;   // 0 or 1
    const int ka   = hi * 2;      // A K-offset: 0 or 2

    v8f acc = {};
    for (int k = 0; k < FEATS; k += 4) {
        v2f a = *(const v2f*)&sX[lrow * FEATS + k + ka];   // ds_load_b64
        v2f b;
        b.x = sW[(k + hi)     * FEATS + n0 + lrow];
        b.y = sW[(k + hi + 2) * FEATS + n0 + lrow];
        acc = __builtin_amdgcn_wmma_f32_16x16x4_f32(false, a, false, b,
                                                    (short)0, acc, false, false);
    }

    const int orow = m0 + hi * 8;
    const int ocol = n0 + lrow;
#pragma unroll
    for (int r = 0; r < 8; ++r) {
        int row = orow + r;
        if (row < n) xw[(size_t)row * FEATS + ocol] = acc[r];
    }
}

__global__ void deg_init(float* deg, int n) {
    int i = blockIdx.x * blockDim.x + threadIdx.x;
    if (i < n) deg[i] = 1.0f;  // self loop
}

__global__ void deg_count(const int* __restrict__ dst, float* deg, int e) {
    int i = blockIdx.x * blockDim.x + threadIdx.x;
    if (i < e) gatomic_add(&deg[dst[i]], 1.0f);
}

__global__ void dinv_k(const float* __restrict__ deg, float* dinv, int n) {
    int i = blockIdx.x * blockDim.x + threadIdx.x;
    if (i < n) dinv[i] = __frsqrt_rn(deg[i]);   // deg >= 1 always (self loop)
}

// h[i,f] = xw[i,f] * dinv[i]^2   (self-loop contribution; also re-initializes
// the accumulator region of d_out every call -> deterministic under replay)
__global__ void self_init(const float* __restrict__ xw, const float* __restrict__ dinv,
                          float* __restrict__ h, int total) {
    int i = blockIdx.x * blockDim.x + threadIdx.x;
    if (i < total) {
        float di = dinv[i >> 7];
        h[i] = xw[i] * di * di;
    }
}

// One wave per edge; each lane owns a float4 of the 128 features.
__global__ __launch_bounds__(256) void edge_scatter(const float* __restrict__ xw,
                                                    const int* __restrict__ src,
                                                    const int* __restrict__ dst,
                                                    const float* __restrict__ dinv,
                                                    float* __restrict__ h, int e) {
    int gw   = (blockIdx.x * 256 + threadIdx.x) >> 5;
    int lane = threadIdx.x & 31;
    if (gw >= e) return;
    int s = src[gw], d = dst[gw];
    float norm = dinv[s] * dinv[d];
    float4 v = ((const float4*)(xw + (size_t)s * FEATS))[lane];   // L2-resident gather
    float* hp = h + (size_t)d * FEATS + lane * 4;
    gatomic_add(hp + 0, v.x * norm);
    gatomic_add(hp + 1, v.y * norm);
    gatomic_add(hp + 2, v.z * norm);
    gatomic_add(hp + 3, v.w * norm);
}

// out1 = h + b (in place over accumulator), out0 = relu(out1)
__global__ void finalize(const float* __restrict__ b, float* __restrict__ out0,
                         float* __restrict__ h, int total) {
    int i = blockIdx.x * blockDim.x + threadIdx.x;
    if (i < total) {
        float v = h[i] + b[i & (FEATS - 1)];
        h[i]    = v;
        out0[i] = fmaxf(v, 0.0f);
    }
}

extern "C" void kernel_launch(void* const* d_in, const int* in_sizes, int n_in,
                              void* d_out, int out_size, void* d_ws, size_t ws_size,
                              hipStream_t stream) {
    const float* x  = (const float*)d_in[0];
    const int*   ei = (const int*)d_in[1];   // edge_index [2, E] row-major int32
    const float* W  = (const float*)d_in[2];
    const float* b  = (const float*)d_in[3];

    const int n = in_sizes[0] / FEATS;
    const int e = in_sizes[1] / 2;
    const int* src = ei;
    const int* dst = ei + e;

    float* out0 = (float*)d_out;                    // relu(h)
    float* h    = out0 + (size_t)n * FEATS;         // pre-relu h; used as accumulator

    float* xw   = (float*)d_ws;                     // n*128 floats
    float* deg  = xw + (size_t)n * FEATS;           // n floats
    float* dinv = deg + n;                          // n floats

    const int total  = n * FEATS;
    const int mtiles = (n + 15) / 16;

    gemm_xw<<<mtiles, 256, 0, stream>>>(x, W, xw, n);
    deg_init<<<(n + 255) / 256, 256, 0, stream>>>(deg, n);
    deg_count<<<(e + 255) / 256, 256, 0, stream>>>(dst, deg, e);
    dinv_k<<<(n + 255) / 256, 256, 0, stream>>>(deg, dinv, n);
    self_init<<<(total + 255) / 256, 256, 0, stream>>>(xw, dinv, h, total);
    edge_scatter<<<(e * 32 + 255) / 256, 256, 0, stream>>>(xw, src, dst, dinv, h, e);
    finalize<<<(total + 255) / 256, 256, 0, stream>>>(b, out0, h, total);
}